// GridMatch_85091892068368
// MI455X (gfx1250) — compile-verified
//
#include <hip/hip_runtime.h>
#include <stdint.h>

// ---------------- problem constants (from reference setup_inputs) ----------
#define A_DIM   8
#define B_DIM   8
#define N_PTS   2048
#define GW_CAP  640
#define GH_CAP  480
#define NBLK    1024                          // blocks for count/emit, width of global scan
#define CELLS_CAP ((size_t)A_DIM * B_DIM * GW_CAP * GH_CAP)   // 19,660,800
#define WORDS_CAP (CELLS_CAP / 32)                            // 614,400

// ---------------- CDNA5 TDM (tensor_load_to_lds) ---------------------------
#if defined(__AMDGCN__) && __has_builtin(__builtin_amdgcn_tensor_load_to_lds) && \
    __has_builtin(__builtin_amdgcn_s_wait_tensorcnt)
#define HAVE_TDM 1
#else
#define HAVE_TDM 0
#endif

#if HAVE_TDM
typedef unsigned int tdm_u32x4 __attribute__((ext_vector_type(4)));
typedef int          tdm_i32x8 __attribute__((ext_vector_type(8)));
typedef int          tdm_i32x4 __attribute__((ext_vector_type(4)));

// 1-D DMA of `nwords` dwords from global -> LDS offset, via Tensor Data Mover.
// D# group0: count=1 | lds_addr | global_addr[56:0] | type=2
// D# group1: data_size=2 (4B), tensor_dim0=tile_dim0=nwords, tensor_dim1=1
static __device__ inline void tdm_load_words(unsigned lds_off, const void* gptr, int nwords) {
  unsigned long long ga = (unsigned long long)(size_t)gptr;
  tdm_u32x4 g0;
  g0.x = 1u;                                            // count=1, user-mode descriptor
  g0.y = lds_off;                                       // lds_addr (bytes)
  g0.z = (unsigned)(ga & 0xFFFFFFFFu);                  // global_addr[31:0]
  g0.w = (unsigned)((ga >> 32) & 0x01FFFFFFu) | (2u << 30); // global_addr[56:32], type=2
  tdm_i32x8 g1;
  g1[0] = 0x00020000;                                   // wg_mask=0, data_size=2 (4 bytes)
  g1[1] = (nwords & 0xFFFF) << 16;                      // tensor_dim0[15:0]
  g1[2] = ((nwords >> 16) & 0xFFFF) | (1 << 16);        // tensor_dim0[31:16], tensor_dim1=1
  g1[3] = (nwords & 0xFFFF) << 16;                      // tensor_dim1 hi=0, tile_dim0=nwords
  g1[4] = 0;                                            // tile_dim1=0, tile_dim2=0 (1-D tile)
  g1[5] = nwords;                                       // tensor_dim0_stride lo
  g1[6] = 0;
  g1[7] = 0;
  tdm_i32x4 z4 = {0, 0, 0, 0};
  tdm_i32x8 z8 = {0, 0, 0, 0, 0, 0, 0, 0};
  // clang-23 / therock-10.0 form: 6 args (g0, g1, g2, g3, extra group, cpol)
  __builtin_amdgcn_tensor_load_to_lds(g0, g1, z4, z4, z8, 0);
}
#endif

// ---------------- helpers ---------------------------------------------------
static __device__ inline bool point_cell(const float* p, int id, int gw, int gh, int* cell) {
  float px = p[2 * id], py = p[2 * id + 1];
  bool vis = (px >= -1.0f) && (px <= 1.0f) && (py >= -1.0f) && (py <= 1.0f);
  if (!vis) return false;
  float sxs = 0.5f * (float)(gw - 1);
  float sys = 0.5f * (float)(gh - 1);
  int x = (int)((px + 1.0f) * sxs);   // truncation == reference astype(int32) (coords >= 0)
  int y = (int)((py + 1.0f) * sys);
  *cell = ((id / N_PTS) * gw + x) * gh + y;   // ((a*B+b)*gw + x)*gh + y
  return true;
}

// ---------------- kernels ---------------------------------------------------
// Defaults: coords -> 0.0, src_bn/dst_an -> -1.0, valid -> 0.0
__global__ void gm_init_out(float* out, int M, long long total) {
  long long i = (long long)blockIdx.x * blockDim.x + threadIdx.x;
  if (i >= total) return;
  long long fourM = 4LL * M, eightM = 8LL * M;
  out[i] = (i < fourM) ? 0.0f : (i < eightM ? -1.0f : 0.0f);
}

// Pass 1: deterministically (re)initialize exactly the grid cells that will be
// written this call. Avoids 157 MB of full-grid memset per call.
__global__ void gm_seed(const float* __restrict__ ps, const float* __restrict__ pd,
                        const int* hp, const int* wp, const int* gp,
                        int* __restrict__ g0, int* __restrict__ g1, int M) {
  int t = blockIdx.x * blockDim.x + threadIdx.x;
  if (t >= 2 * M) return;
  int gs = gp[0];
  int gh = hp[0] * gs, gw = wp[0] * gs;
  const float* p = (t < M) ? ps : pd;
  int* g = (t < M) ? g0 : g1;
  int id = (t < M) ? t : t - M;
  int cell;
  if (point_cell(p, id, gw, gh, &cell)) g[cell] = -1;
}

// Pass 2: scatter-max of point id (== last-write-wins over ascending arange)
// and set occupancy bit.
__global__ void gm_scatter(const float* __restrict__ ps, const float* __restrict__ pd,
                           const int* hp, const int* wp, const int* gp,
                           int* __restrict__ g0, int* __restrict__ g1,
                           unsigned* __restrict__ b0, unsigned* __restrict__ b1, int M) {
  int t = blockIdx.x * blockDim.x + threadIdx.x;
  if (t >= 2 * M) return;
  int gs = gp[0];
  int gh = hp[0] * gs, gw = wp[0] * gs;
  const float* p = (t < M) ? ps : pd;
  int* g = (t < M) ? g0 : g1;
  unsigned* bits = (t < M) ? b0 : b1;
  int id = (t < M) ? t : t - M;
  int cell;
  if (point_cell(p, id, gw, gh, &cell)) {
    atomicMax(&g[cell], id);
    atomicOr(&bits[cell >> 5], 1u << (cell & 31));
  }
}

// Pass 3: per-block match counts over bitmask words (reads 5 MB, not 157 MB).
__global__ void gm_count(const unsigned* __restrict__ b0, const unsigned* __restrict__ b1,
                         const int* hp, const int* wp, const int* gp,
                         int* __restrict__ bcnt) {
  __shared__ int sred[256];
  int gs = gp[0];
  int gh = hp[0] * gs, gw = wp[0] * gs;
  long long cells = (long long)A_DIM * B_DIM * gw * gh;
  int words = (int)((cells + 31) >> 5);
  int chunk = (words + NBLK - 1) / NBLK;
  int start = blockIdx.x * chunk;
  int endw = start + chunk; if (endw > words) endw = words;
  int c = 0;
  for (int w = start + threadIdx.x; w < endw; w += blockDim.x)
    c += __popc(b0[w] & b1[w]);
  sred[threadIdx.x] = c;
  __syncthreads();
  for (int off = 128; off > 0; off >>= 1) {
    if ((int)threadIdx.x < off) sred[threadIdx.x] += sred[threadIdx.x + off];
    __syncthreads();
  }
  if (threadIdx.x == 0) bcnt[blockIdx.x] = sred[0];
}

// Pass 4: exclusive scan of NBLK block counts (single block, LDS Hillis-Steele).
__global__ void gm_scan(const int* __restrict__ bcnt, int* __restrict__ boff) {
  __shared__ int s[NBLK];
  int t = threadIdx.x;
  int own = bcnt[t];
  s[t] = own;
  __syncthreads();
  for (int off = 1; off < NBLK; off <<= 1) {
    int v = (t >= off) ? s[t - off] : 0;
    __syncthreads();
    s[t] += v;
    __syncthreads();
  }
  boff[t] = s[t] - own;
  if (t == NBLK - 1) boff[NBLK] = s[t];
}

// Pass 5: ordered emit. Bitmask chunk staged into LDS via TDM (async tensor
// DMA), then tile-wise 256-wide LDS exclusive scans preserve row-major order.
__global__ void gm_emit(const float* __restrict__ ps, const float* __restrict__ pd,
                        const int* hp, const int* wp, const int* gp,
                        const int* __restrict__ g0, const int* __restrict__ g1,
                        const unsigned* __restrict__ b0, const unsigned* __restrict__ b1,
                        const int* __restrict__ boff, float* __restrict__ out, int M) {
  __shared__ unsigned sb0[1024];
  __shared__ unsigned sb1[1024];
  __shared__ int sscan[256];

  int gs = gp[0];
  int h = hp[0], w_ = wp[0];
  int gh = h * gs, gw = w_ * gs;
  long long cells = (long long)A_DIM * B_DIM * gw * gh;
  int words = (int)((cells + 31) >> 5);
  int chunk = (words + NBLK - 1) / NBLK;
  int start = blockIdx.x * chunk;
  int bw = words - start;
  if (bw > chunk) bw = chunk;
  if (bw < 0) bw = 0;

  bool useLds = false;
#if HAVE_TDM
  useLds = (bw > 0 && bw <= 1024);
  if (useLds && threadIdx.x == 0) {
    tdm_load_words((unsigned)(size_t)&sb0[0], b0 + start, bw);
    tdm_load_words((unsigned)(size_t)&sb1[0], b1 + start, bw);
    __builtin_amdgcn_s_wait_tensorcnt(0);
  }
  __syncthreads();
#endif

  float sxs = 0.5f * (float)(gw - 1);
  float sys = 0.5f * (float)(gh - 1);
  float fx = (float)(h - 1) / (float)(gh - 1);   // faithful to source: pairs with x
  float fy = (float)(w_ - 1) / (float)(gw - 1);  // pairs with y

  int base = boff[blockIdx.x];
  long long oD = 2LL * M, oB = 4LL * M, oA = 6LL * M, oV = 8LL * M;

  for (int tb = 0; tb < bw; tb += 256) {
    int wi = tb + (int)threadIdx.x;
    unsigned mask = 0;
    if (wi < bw) mask = useLds ? (sb0[wi] & sb1[wi]) : (b0[start + wi] & b1[start + wi]);
    int cnt = __popc(mask);

    sscan[threadIdx.x] = cnt;
    __syncthreads();
    for (int off = 1; off < 256; off <<= 1) {
      int v = ((int)threadIdx.x >= off) ? sscan[threadIdx.x - off] : 0;
      __syncthreads();
      sscan[threadIdx.x] += v;
      __syncthreads();
    }
    int excl = sscan[threadIdx.x] - cnt;
    int tileTotal = sscan[255];

    int r = base + excl;
    unsigned m = mask;
    while (m) {
      int bit = __builtin_ctz(m);
      m &= m - 1;
      int cell = ((start + wi) << 5) + bit;
      int si = g0[cell];
      int di = g1[cell];
      float spx = ps[2 * si], spy = ps[2 * si + 1];
      float dpx = pd[2 * di], dpy = pd[2 * di + 1];
      out[2 * r]          = (spx + 1.0f) * sxs * fx;
      out[2 * r + 1]      = (spy + 1.0f) * sys * fy;
      out[oD + 2 * r]     = (dpx + 1.0f) * sxs * fx;
      out[oD + 2 * r + 1] = (dpy + 1.0f) * sys * fy;
      out[oB + 2 * r]     = (float)((si / N_PTS) % B_DIM);
      out[oB + 2 * r + 1] = (float)(si % N_PTS);
      out[oA + 2 * r]     = (float)(di / (B_DIM * N_PTS));
      out[oA + 2 * r + 1] = (float)(di % N_PTS);
      out[oV + r]         = 1.0f;
      r++;
    }
    base += tileTotal;
    __syncthreads();   // protect sscan before next tile overwrites it
  }
}

// ---------------- host launcher --------------------------------------------
extern "C" void kernel_launch(void* const* d_in, const int* in_sizes, int n_in,
                              void* d_out, int out_size, void* d_ws, size_t ws_size,
                              hipStream_t stream) {
  const float* p_src = (const float*)d_in[0];
  const float* p_dst = (const float*)d_in[1];
  const int* hp = (const int*)d_in[2];
  const int* wp = (const int*)d_in[3];
  const int* gp = (const int*)d_in[4];
  float* out = (float*)d_out;

  const int M = in_sizes[0] / 2;        // A*B*N points (131072)

  // workspace layout
  char* ws = (char*)d_ws;
  int*      grid0 = (int*)ws;                                   // CELLS_CAP int32
  int*      grid1 = grid0 + CELLS_CAP;                          // CELLS_CAP int32
  unsigned* bits0 = (unsigned*)(grid1 + CELLS_CAP);             // WORDS_CAP u32
  unsigned* bits1 = bits0 + WORDS_CAP;                          // WORDS_CAP u32
  int*      bcnt  = (int*)(bits1 + WORDS_CAP);                  // NBLK int32
  int*      boff  = bcnt + NBLK;                                // NBLK+1 int32

  // clear occupancy bitmasks only (5 MB, not the 157 MB value grids)
  (void)hipMemsetAsync(bits0, 0, 2 * WORDS_CAP * sizeof(unsigned), stream);

  long long totalOut = 9LL * M;
  if (totalOut > (long long)out_size) totalOut = out_size;
  int initBlocks = (int)((totalOut + 255) / 256);
  gm_init_out<<<initBlocks, 256, 0, stream>>>(out, M, totalOut);

  int ptBlocks = (2 * M + 255) / 256;
  gm_seed<<<ptBlocks, 256, 0, stream>>>(p_src, p_dst, hp, wp, gp, grid0, grid1, M);
  gm_scatter<<<ptBlocks, 256, 0, stream>>>(p_src, p_dst, hp, wp, gp,
                                           grid0, grid1, bits0, bits1, M);
  gm_count<<<NBLK, 256, 0, stream>>>(bits0, bits1, hp, wp, gp, bcnt);
  gm_scan<<<1, NBLK, 0, stream>>>(bcnt, boff);
  gm_emit<<<NBLK, 256, 0, stream>>>(p_src, p_dst, hp, wp, gp,
                                    grid0, grid1, bits0, bits1, boff, out, M);
}